// N2N_558345748713
// MI455X (gfx1250) — compile-verified
//
#include <hip/hip_runtime.h>
#include <math.h>

#define B_  32
#define M_  100
#define L_  30
#define LQ_ 30
#define E_  256
#define V_  32000

typedef __attribute__((ext_vector_type(2))) float v2f;
typedef __attribute__((ext_vector_type(8))) float v8f;

// ---------------------------------------------------------------------------
// K1: query embedding  u0[b][e] = sum_j A1[Q[b][j]][e] * pe_q(j,e) * QM[b][j]
// ---------------------------------------------------------------------------
__global__ void query_embed_kernel(const int* __restrict__ trainQ,
                                   const float* __restrict__ trainQM,
                                   const float* __restrict__ A1,
                                   float* __restrict__ u0) {
  int b = blockIdx.x;
  int e = threadIdx.x;
  const float half_e = (E_ + 1) * 0.5f;
  const float half_j = (LQ_ + 1) * 0.5f;
  const float scale  = 4.0f / (float)(E_ * LQ_);
  float ke = ((float)(e + 1) - half_e) * scale;
  float acc = 0.f;
  for (int j = 0; j < LQ_; ++j) {
    int idx = trainQ[b * LQ_ + j];
    float pe = 1.0f + ke * ((float)(j + 1) - half_j);
    acc += A1[(size_t)idx * E_ + e] * pe * trainQM[b * LQ_ + j];
  }
  u0[b * E_ + e] = acc;
}

// ---------------------------------------------------------------------------
// K2: story embeddings (the bandwidth-dominant gather, ~393MB)
// emb layout: [4][B*M][E]   embX[bm][e] = sum_l AX[S[bm][l]][e] * pe_s(l,e)
// ---------------------------------------------------------------------------
__global__ void story_embed_kernel(const int* __restrict__ trainS,
                                   const float* __restrict__ A1,
                                   const float* __restrict__ A2,
                                   const float* __restrict__ A3,
                                   const float* __restrict__ A4,
                                   float* __restrict__ emb) {
  int bm = blockIdx.x;
  int e  = threadIdx.x;
  __shared__ int idx_s[L_];
  if (threadIdx.x < L_) idx_s[threadIdx.x] = trainS[bm * L_ + threadIdx.x];
  __syncthreads();
  const float half_e = (E_ + 1) * 0.5f;
  const float half_j = (L_ + 1) * 0.5f;
  const float scale  = 4.0f / (float)(E_ * L_);
  float ke = ((float)(e + 1) - half_e) * scale;
  float a1 = 0.f, a2 = 0.f, a3 = 0.f, a4 = 0.f;
  for (int l = 0; l < L_; ++l) {
    size_t off = (size_t)idx_s[l] * E_ + e;
    if (l + 1 < L_) {
      size_t offn = (size_t)idx_s[l + 1] * E_ + e;
      __builtin_prefetch(&A1[offn], 0, 0);   // global_prefetch_b8
      __builtin_prefetch(&A2[offn], 0, 0);
      __builtin_prefetch(&A3[offn], 0, 0);
      __builtin_prefetch(&A4[offn], 0, 0);
    }
    float pe = 1.0f + ke * ((float)(l + 1) - half_j);
    a1 += A1[off] * pe;
    a2 += A2[off] * pe;
    a3 += A3[off] * pe;
    a4 += A4[off] * pe;
  }
  size_t base = (size_t)bm * E_ + e;
  const size_t stride = (size_t)B_ * M_ * E_;
  emb[base]              = a1;
  emb[base + stride]     = a2;
  emb[base + 2 * stride] = a3;
  emb[base + 3 * stride] = a4;
}

// ---------------------------------------------------------------------------
// K3: three memory hops, one block per batch element, u kept in LDS
// ---------------------------------------------------------------------------
__global__ void hops_kernel(const float* __restrict__ u0,
                            const float* __restrict__ emb,
                            const float* __restrict__ trainPM,
                            float* __restrict__ uF) {
  int b = blockIdx.x;
  int t = threadIdx.x;
  int lane = t & 31, wv = t >> 5;
  __shared__ float u_s[E_];
  __shared__ float sc_s[M_];
  __shared__ float p_s[M_];
  u_s[t] = u0[b * E_ + t];
  __syncthreads();
  const size_t stride = (size_t)B_ * M_ * E_;
  for (int hop = 0; hop < 3; ++hop) {
    const float* memA = emb + (size_t)hop * stride + (size_t)b * M_ * E_;
    const float* memC = memA + stride;
    // scores[m] = (memA[m] . u) * PM[b][m]   (wave-per-m, shuffle reduce)
    for (int m = wv; m < M_; m += 8) {
      const float* row = memA + (size_t)m * E_;
      float s = 0.f;
      #pragma unroll
      for (int e = lane; e < E_; e += 32) s += row[e] * u_s[e];
      #pragma unroll
      for (int off = 16; off > 0; off >>= 1) s += __shfl_xor(s, off, 32);
      if (lane == 0) sc_s[m] = s * trainPM[b * M_ + m];
    }
    __syncthreads();
    // masked softmax over M=100 in a single wave32
    if (t < 32) {
      float xm[4], pmv[4], ev[4];
      float mx = -INFINITY;
      #pragma unroll
      for (int i = 0; i < 4; ++i) {
        int m = lane + 32 * i;
        if (m < M_) {
          pmv[i] = trainPM[b * M_ + m];
          xm[i]  = (pmv[i] > 0.f) ? sc_s[m] : -1e30f;
          mx = fmaxf(mx, xm[i]);
        } else { pmv[i] = 0.f; xm[i] = -INFINITY; }
      }
      #pragma unroll
      for (int off = 16; off > 0; off >>= 1) mx = fmaxf(mx, __shfl_xor(mx, off, 32));
      float sum = 0.f;
      #pragma unroll
      for (int i = 0; i < 4; ++i) {
        int m = lane + 32 * i;
        ev[i] = (m < M_) ? expf(xm[i] - mx) * pmv[i] : 0.f;
        sum += ev[i];
      }
      #pragma unroll
      for (int off = 16; off > 0; off >>= 1) sum += __shfl_xor(sum, off, 32);
      float inv = 1.0f / (sum + 1e-13f);
      #pragma unroll
      for (int i = 0; i < 4; ++i) {
        int m = lane + 32 * i;
        if (m < M_) p_s[m] = ev[i] * inv;
      }
    }
    __syncthreads();
    // o[e] = sum_m p[m] * memC[m][e];  u = u + o   (coalesced over e)
    float uv  = u_s[t];
    float acc = 0.f;
    for (int m = 0; m < M_; ++m) acc += p_s[m] * memC[(size_t)m * E_ + t];
    __syncthreads();
    u_s[t] = uv + acc;
    __syncthreads();
  }
  uF[b * E_ + t] = u_s[t];
}

// ---------------------------------------------------------------------------
// K4a: wx = u @ W.T + bias via V_WMMA_F32_16X16X4_F32, fused batch-norm
// (axis-0 over B=32 lives inside one wave tile -> one shuffle) + mask shift.
// Each wave owns a 32x16 v-tile (two 16x16 accumulators); block covers 128 v.
// ---------------------------------------------------------------------------
__global__ void gemm_bn_kernel(const float* __restrict__ uF,
                               const float* __restrict__ Wm,
                               const float* __restrict__ bias,
                               const float* __restrict__ gamma,
                               const float* __restrict__ beta,
                               const float* __restrict__ VM,
                               float* __restrict__ z) {
  __shared__ float u_s[B_ * E_];           // 32 KB
  int t = threadIdx.x;
  #pragma unroll
  for (int i = 0; i < (B_ * E_) / 256; ++i) u_s[t + i * 256] = uF[t + i * 256];
  __syncthreads();

  int lane = t & 31, wv = t >> 5;
  int n  = lane & 15;                      // column-in-tile / M-row-in-tile
  int hi = lane >> 4;                      // lane half selects K pair / row half
  int kb = hi << 1;
  int v  = blockIdx.x * 128 + wv * 16 + n;
  const float* Wrow = Wm + (size_t)v * E_;

  v8f acc0 = {};                           // rows 0..15 of wx tile
  v8f acc1 = {};                           // rows 16..31
  for (int e0 = 0; e0 < E_; e0 += 4) {
    v2f a_lo, a_hi, bv;
    // A(16x4) layout: lane = M + 16*(K>=2), vgpr = K&1
    a_lo.x = u_s[n * E_ + e0 + kb];
    a_lo.y = u_s[n * E_ + e0 + kb + 1];
    a_hi.x = u_s[(n + 16) * E_ + e0 + kb];
    a_hi.y = u_s[(n + 16) * E_ + e0 + kb + 1];
    // B(4x16) layout mirrors with N; B[k][n] = W[v][e0+k]
    bv.x = Wrow[e0 + kb];
    bv.y = Wrow[e0 + kb + 1];
    acc0 = __builtin_amdgcn_wmma_f32_16x16x4_f32(false, a_lo, false, bv,
                                                 (short)0, acc0, false, false);
    acc1 = __builtin_amdgcn_wmma_f32_16x16x4_f32(false, a_hi, false, bv,
                                                 (short)0, acc1, false, false);
  }

  // add bias, per-column (over B=32) mean/var: partial in-lane, then xor-16
  float bb = bias[v];
  float ps = 0.f, ps2 = 0.f;
  #pragma unroll
  for (int r = 0; r < 8; ++r) {
    float x0 = acc0[r] + bb, x1 = acc1[r] + bb;
    acc0[r] = x0; acc1[r] = x1;
    ps  += x0 + x1;
    ps2 += x0 * x0 + x1 * x1;
  }
  ps  += __shfl_xor(ps, 16, 32);
  ps2 += __shfl_xor(ps2, 16, 32);
  float mean = ps * (1.0f / 32.0f);
  float var  = ps2 * (1.0f / 32.0f) - mean * mean;
  float inv  = rsqrtf(var + 1e-5f);
  float g = gamma[v], be = beta[v];

  // C/D layout: vgpr r holds row r (lanes 0-15) and row r+8 (lanes 16-31)
  #pragma unroll
  for (int r = 0; r < 8; ++r) {
    int row0 = r + 8 * hi;
    int row1 = row0 + 16;
    float y0 = g * (acc0[r] - mean) * inv + be;
    float y1 = g * (acc1[r] - mean) * inv + be;
    z[(size_t)row0 * V_ + v] = y0 + logf(VM[(size_t)row0 * V_ + v] + 1e-13f);
    z[(size_t)row1 * V_ + v] = y1 + logf(VM[(size_t)row1 * V_ + v] + 1e-13f);
  }
}

// ---------------------------------------------------------------------------
// K4b: per-row max and log-sum-exp over V (deterministic block reductions)
// ---------------------------------------------------------------------------
__global__ void rowstats_kernel(const float* __restrict__ z,
                                float* __restrict__ stats) {
  int b = blockIdx.x;
  int t = threadIdx.x;
  __shared__ float red[256];
  const float* row = z + (size_t)b * V_;
  float mx = -INFINITY;
  for (int v = t; v < V_; v += 256) mx = fmaxf(mx, row[v]);
  red[t] = mx; __syncthreads();
  for (int s = 128; s > 0; s >>= 1) {
    if (t < s) red[t] = fmaxf(red[t], red[t + s]);
    __syncthreads();
  }
  mx = red[0];
  __syncthreads();
  float sum = 0.f;
  for (int v = t; v < V_; v += 256) sum += expf(row[v] - mx);
  red[t] = sum; __syncthreads();
  for (int s = 128; s > 0; s >>= 1) {
    if (t < s) red[t] += red[t + s];
    __syncthreads();
  }
  if (t == 0) { stats[b * 2] = mx; stats[b * 2 + 1] = logf(red[0]); }
}

// ---------------------------------------------------------------------------
// K4c: out = z - rowmax - log(rowsum)   (in place on d_out)
// ---------------------------------------------------------------------------
__global__ void finalize_kernel(float* __restrict__ z,
                                const float* __restrict__ stats) {
  int i = blockIdx.x * 256 + threadIdx.x;
  if (i < B_ * V_) {
    int b = i / V_;
    z[i] = z[i] - stats[b * 2] - stats[b * 2 + 1];
  }
}

extern "C" void kernel_launch(void* const* d_in, const int* in_sizes, int n_in,
                              void* d_out, int out_size, void* d_ws, size_t ws_size,
                              hipStream_t stream) {
  const int*   trainS  = (const int*)d_in[0];
  const int*   trainQ  = (const int*)d_in[1];
  const float* trainVM = (const float*)d_in[2];
  const float* trainPM = (const float*)d_in[3];
  // d_in[4] = trainSM (unused by the reference math)
  const float* trainQM = (const float*)d_in[5];
  const float* A1      = (const float*)d_in[6];
  const float* A2      = (const float*)d_in[7];
  const float* A3      = (const float*)d_in[8];
  const float* A4      = (const float*)d_in[9];
  const float* Wm      = (const float*)d_in[10];
  const float* bias    = (const float*)d_in[11];
  const float* gamma   = (const float*)d_in[12];
  const float* beta    = (const float*)d_in[13];

  float* ws    = (float*)d_ws;
  float* u0    = ws;                                   // B*E
  float* emb   = u0 + (size_t)B_ * E_;                 // 4*B*M*E
  float* uF    = emb + 4 * (size_t)B_ * M_ * E_;       // B*E
  float* stats = uF + (size_t)B_ * E_;                 // 2*B
  float* z     = (float*)d_out;                        // B*V

  query_embed_kernel<<<B_, E_, 0, stream>>>(trainQ, trainQM, A1, u0);
  story_embed_kernel<<<B_ * M_, E_, 0, stream>>>(trainS, A1, A2, A3, A4, emb);
  hops_kernel<<<B_, E_, 0, stream>>>(u0, emb, trainPM, uF);
  gemm_bn_kernel<<<V_ / 128, 256, 0, stream>>>(uF, Wm, bias, gamma, beta, trainVM, z);
  rowstats_kernel<<<B_, 256, 0, stream>>>(z, stats);
  finalize_kernel<<<(B_ * V_ + 255) / 256, 256, 0, stream>>>(z, stats);
}